// NeuralDiffusionEngine_85212151153350
// MI455X (gfx1250) — compile-verified
//
#include <hip/hip_runtime.h>
#include <math.h>
#include <stdint.h>

// Problem geometry (D is compile-time: needed for LDS sizing & circular masks)
constexpr int kD    = 2048;
constexpr int kMask = 2047;
constexpr int kNT   = 256;   // threads per row-block (8 wave32 waves)
constexpr int kEPT  = 8;     // contiguous elements per thread (256*8 = 2048)

// ---------------------------------------------------------------------------
// CDNA5 helpers
// ---------------------------------------------------------------------------

// Low 32 bits of a flat pointer to LDS == workgroup-relative LDS byte address.
__device__ __forceinline__ uint32_t lds_addr_of(const void* p) {
  return (uint32_t)(uintptr_t)p;
}

// Async global -> LDS copy, 16 bytes per active lane (GVS mode: SGPR base +
// per-lane VGPR byte offset). Tracked by ASYNCcnt.
__device__ __forceinline__ void async_load_b128(uint32_t lds, uint32_t off, const void* base) {
  asm volatile("global_load_async_to_lds_b128 %0, %1, %2"
               :: "v"(lds), "v"(off), "s"(base) : "memory");
}
// Same, with non-temporal hint (streamed data we never reuse: psi).
__device__ __forceinline__ void async_load_b128_nt(uint32_t lds, uint32_t off, const void* base) {
  asm volatile("global_load_async_to_lds_b128 %0, %1, %2 th:TH_LOAD_NT"
               :: "v"(lds), "v"(off), "s"(base) : "memory");
}
__device__ __forceinline__ void wait_async0() {
  asm volatile("s_wait_asynccnt 0" ::: "memory");
}

// wave32 butterfly via ds_swizzle (group-of-32 mode: and=0x1f, or=0, xor=M)
template <int M>
__device__ __forceinline__ float swz_xor(float v) {
  return __int_as_float(__builtin_amdgcn_ds_swizzle(__float_as_int(v), (M << 10) | 0x1f));
}
__device__ __forceinline__ float wave_sum(float v) {
  v += swz_xor<1>(v); v += swz_xor<2>(v); v += swz_xor<4>(v);
  v += swz_xor<8>(v); v += swz_xor<16>(v);
  return v;
}
__device__ __forceinline__ float wave_max(float v) {
  v = fmaxf(v, swz_xor<1>(v)); v = fmaxf(v, swz_xor<2>(v)); v = fmaxf(v, swz_xor<4>(v));
  v = fmaxf(v, swz_xor<8>(v)); v = fmaxf(v, swz_xor<16>(v));
  return v;
}
__device__ __forceinline__ float wave_min(float v) {
  v = fminf(v, swz_xor<1>(v)); v = fminf(v, swz_xor<2>(v)); v = fminf(v, swz_xor<4>(v));
  v = fminf(v, swz_xor<8>(v)); v = fminf(v, swz_xor<16>(v));
  return v;
}

__device__ __forceinline__ float sigmoidf(float x) { return 1.0f / (1.0f + expf(-x)); }

// ---------------------------------------------------------------------------
// Pass 1: per-row statistics.  One block per row; row staged in LDS async.
// Thread t owns contiguous chunk [8t, 8t+8); gradients from registers + 2 halo
// LDS reads.  Outputs per row: dispersion, tot, min|f|, max|f|.
// ---------------------------------------------------------------------------
__global__ void __launch_bounds__(kNT)
rowstats_kernel(const float* __restrict__ psi, const float* __restrict__ field,
                float* __restrict__ disp, float* __restrict__ tot,
                float* __restrict__ fmn, float* __restrict__ fmx) {
  __shared__ __align__(16) float ps[kD];
  __shared__ __align__(16) float fs[kD];
  __shared__ float red[8][9];

  const int b = blockIdx.x;
  const int t = threadIdx.x;
  const size_t rowoff = (size_t)b * kD;

  const uint32_t ps_a = lds_addr_of(ps);
  const uint32_t fs_a = lds_addr_of(fs);
#pragma unroll
  for (int c = 0; c < 2; ++c) {
    const uint32_t off = (uint32_t)(t * 32 + c * 16);
    async_load_b128_nt(ps_a + off, off, psi + rowoff);   // psi read once -> NT
    async_load_b128(fs_a + off, off, field + rowoff);    // field reused in pass 3 -> RT
  }
  wait_async0();
  __syncthreads();

  const int i0 = t * kEPT;
  // x[0] = left halo (i0-1), x[1..8] = chunk, x[9] = right halo (i0+8)
  float xp[kEPT + 2], xf[kEPT + 2];
  {
    const float4* p4 = (const float4*)ps;
    const float4* f4 = (const float4*)fs;
    float4 a = p4[2 * t], c = p4[2 * t + 1];
    xp[1] = a.x; xp[2] = a.y; xp[3] = a.z; xp[4] = a.w;
    xp[5] = c.x; xp[6] = c.y; xp[7] = c.z; xp[8] = c.w;
    float4 e = f4[2 * t], g = f4[2 * t + 1];
    xf[1] = e.x; xf[2] = e.y; xf[3] = e.z; xf[4] = e.w;
    xf[5] = g.x; xf[6] = g.y; xf[7] = g.z; xf[8] = g.w;
    xp[0] = ps[(i0 - 1) & kMask]; xp[kEPT + 1] = ps[(i0 + kEPT) & kMask];
    xf[0] = fs[(i0 - 1) & kMask]; xf[kEPT + 1] = fs[(i0 + kEPT) & kMask];
  }

  float s1 = 0.f, s2 = 0.f, se = 0.f;
  float sgp = 0.f, sgp2 = 0.f, sgf = 0.f, sgf2 = 0.f;
  float mn = 3.0e38f, mx = 0.0f;
#pragma unroll
  for (int j = 0; j < kEPT; ++j) {
    const float x = xp[j + 1];
    s1 += x;
    s2 += x * x;
    se += x * logf(x + 1e-10f);
    const float af = fabsf(xf[j + 1]);
    mn = fminf(mn, af);
    mx = fmaxf(mx, af);
    const float gp = 0.5f * (xp[j + 2] - xp[j]);
    const float gf = 0.5f * (xf[j + 2] - xf[j]);
    sgp += gp; sgp2 += gp * gp;
    sgf += gf; sgf2 += gf * gf;
  }
  s1 = wave_sum(s1);   s2 = wave_sum(s2);   se = wave_sum(se);
  sgp = wave_sum(sgp); sgp2 = wave_sum(sgp2);
  sgf = wave_sum(sgf); sgf2 = wave_sum(sgf2);
  mn = wave_min(mn);   mx = wave_max(mx);

  const int wid = t >> 5, lid = t & 31;
  if (lid == 0) {
    red[wid][0] = s1;  red[wid][1] = s2;  red[wid][2] = se;
    red[wid][3] = sgp; red[wid][4] = sgp2;
    red[wid][5] = sgf; red[wid][6] = sgf2;
    red[wid][7] = mn;  red[wid][8] = mx;
  }
  __syncthreads();
  if (t == 0) {
    float S1 = 0, S2 = 0, SE = 0, SGP = 0, SGP2 = 0, SGF = 0, SGF2 = 0;
    float MN = 3.0e38f, MX = 0.0f;
#pragma unroll
    for (int wv = 0; wv < 8; ++wv) {
      S1 += red[wv][0]; S2 += red[wv][1]; SE += red[wv][2];
      SGP += red[wv][3]; SGP2 += red[wv][4];
      SGF += red[wv][5]; SGF2 += red[wv][6];
      MN = fminf(MN, red[wv][7]); MX = fmaxf(MX, red[wv][8]);
    }
    const float invD  = 1.0f / (float)kD;
    const float invN1 = 1.0f / (float)(kD - 1);
    const float nent  = (-SE) * (1.0f / 7.6246189861593985f);  // entropy / log(2048)
    const float var   = fmaxf(0.0f, (S2 - S1 * S1 * invD) * invN1);   // ddof=1
    const float dispv = fminf(fmaxf(0.7f * nent + 0.3f * sqrtf(var), 0.0f), 1.0f);
    const float vgp   = fmaxf(0.0f, (SGP2 - SGP * SGP * invD) * invN1);
    const float vgf   = fmaxf(0.0f, (SGF2 - SGF * SGF * invD) * invN1);
    disp[b] = dispv;
    tot[b]  = sqrtf(vgp + vgf);   // sqrt(pc^2 + fc^2)
    fmn[b]  = MN;
    fmx[b]  = MX;
  }
}

// ---------------------------------------------------------------------------
// Pass 2: global scalars + per-row constants. Single block of 1024 threads.
// combined[b,i] = rowc[b] + w1*sigmoid(5*(|f|/(gmaxf+eps)-0.5)); since sigmoid
// is monotone, global min/max of combined come from per-row min/max of |f|.
// ---------------------------------------------------------------------------
__global__ void __launch_bounds__(1024)
globalred_kernel(const float* __restrict__ w,
                 const float* __restrict__ disp, const float* __restrict__ tot,
                 const float* __restrict__ fmn, const float* __restrict__ fmx,
                 float* __restrict__ rowc, float* __restrict__ scalars, int nrows) {
  __shared__ float lred[32];
  __shared__ float s_gmaxf, s_gmaxt;
  const int t = threadIdx.x;
  const int wid = t >> 5, lid = t & 31;

  float mf = 0.0f, mt = 0.0f;
  for (int b = t; b < nrows; b += 1024) {
    mf = fmaxf(mf, fmx[b]);
    mt = fmaxf(mt, tot[b]);
  }
  mf = wave_max(mf);
  if (lid == 0) lred[wid] = mf;
  __syncthreads();
  if (wid == 0) {
    float v = wave_max(lred[lid]);
    if (lid == 0) s_gmaxf = v;
  }
  __syncthreads();
  mt = wave_max(mt);
  if (lid == 0) lred[wid] = mt;
  __syncthreads();
  if (wid == 0) {
    float v = wave_max(lred[lid]);
    if (lid == 0) s_gmaxt = v;
  }
  __syncthreads();

  const float gmaxf = s_gmaxf, gmaxt = s_gmaxt;
  const float invg = 1.0f / (gmaxf + 1e-10f);
  const float invt = 1.0f / (gmaxt + 1e-10f);
  const float w0 = w[0], w1 = w[1], w2 = w[2], w3 = w[3];

  float cmin = 3.0e38f, cmax = -3.0e38f;
  for (int b = t; b < nrows; b += 1024) {
    const float comp = fminf(fmaxf(tot[b] * invt, 0.0f), 1.0f);
    const float rc = w0 * disp[b] + w2 * comp + w3 * 0.5f;
    rowc[b] = rc;
    const float fl = w1 * sigmoidf(5.0f * (fmn[b] * invg - 0.5f));
    const float fh = w1 * sigmoidf(5.0f * (fmx[b] * invg - 0.5f));
    cmin = fminf(cmin, rc + fminf(fl, fh));   // sign-of-w1 safe
    cmax = fmaxf(cmax, rc + fmaxf(fl, fh));
  }
  __syncthreads();
  cmin = wave_min(cmin);
  if (lid == 0) lred[wid] = cmin;
  __syncthreads();
  if (wid == 0) {
    float v = wave_min(lred[lid]);
    if (lid == 0) scalars[1] = v;
  }
  __syncthreads();
  cmax = wave_max(cmax);
  if (lid == 0) lred[wid] = cmax;
  __syncthreads();
  if (wid == 0) {
    float v = wave_max(lred[lid]);
    if (lid == 0) {
      scalars[2] = v;
      scalars[0] = gmaxf;
      scalars[3] = gmaxt;
    }
  }
}

// ---------------------------------------------------------------------------
// Pass 3: elementwise dc + circular [0.25,0.5,0.25] smoothing.  dc kept in
// registers; LDS only provides the two halo values per thread.  Output is
// written as two b128 stores per thread.  field should hit L2 from pass 1.
// ---------------------------------------------------------------------------
__global__ void __launch_bounds__(kNT)
finalize_kernel(const float* __restrict__ field, const float* __restrict__ w,
                const float* __restrict__ rowc, const float* __restrict__ scalars,
                float* __restrict__ out) {
  __shared__ __align__(16) float fs[kD];
  __shared__ __align__(16) float dc[kD];
  const int b = blockIdx.x;
  const int t = threadIdx.x;
  const size_t rowoff = (size_t)b * kD;

  const uint32_t fs_a = lds_addr_of(fs);
#pragma unroll
  for (int c = 0; c < 2; ++c) {
    const uint32_t off = (uint32_t)(t * 32 + c * 16);
    async_load_b128(fs_a + off, off, field + rowoff);
  }
  // uniform scalar loads overlap with the async copy
  const float invg  = 1.0f / (scalars[0] + 1e-10f);
  const float cmin  = scalars[1];
  const float scale = 0.099f / (scalars[2] - cmin + 1e-10f);  // (D_MAX-D_MIN)/range
  const float w1 = w[1];
  const float rc = rowc[b];
  wait_async0();
  __syncthreads();

  const int i0 = t * kEPT;
  float f[kEPT], d[kEPT];
  {
    const float4* f4 = (const float4*)fs;
    float4 a = f4[2 * t], c = f4[2 * t + 1];
    f[0] = a.x; f[1] = a.y; f[2] = a.z; f[3] = a.w;
    f[4] = c.x; f[5] = c.y; f[6] = c.z; f[7] = c.w;
  }
#pragma unroll
  for (int j = 0; j < kEPT; ++j) {
    const float af = fabsf(f[j]);
    const float ff = sigmoidf(5.0f * (af * invg - 0.5f));
    const float comb = rc + w1 * ff;
    d[j] = 0.001f + (comb - cmin) * scale;
  }
  {
    float4* d4 = (float4*)dc;
    d4[2 * t]     = make_float4(d[0], d[1], d[2], d[3]);
    d4[2 * t + 1] = make_float4(d[4], d[5], d[6], d[7]);
  }
  __syncthreads();
  const float dm = dc[(i0 - 1) & kMask];        // left halo
  const float dp = dc[(i0 + kEPT) & kMask];     // right halo

  float sm[kEPT];
#pragma unroll
  for (int j = 0; j < kEPT; ++j) {
    const float left  = (j == 0)        ? dm : d[j - 1];
    const float right = (j == kEPT - 1) ? dp : d[j + 1];
    sm[j] = 0.25f * left + 0.5f * d[j] + 0.25f * right;
  }
  float4* out4 = (float4*)(out + rowoff);
  out4[2 * t]     = make_float4(sm[0], sm[1], sm[2], sm[3]);
  out4[2 * t + 1] = make_float4(sm[4], sm[5], sm[6], sm[7]);
}

// ---------------------------------------------------------------------------
extern "C" void kernel_launch(void* const* d_in, const int* in_sizes, int n_in,
                              void* d_out, int out_size, void* d_ws, size_t ws_size,
                              hipStream_t stream) {
  (void)n_in; (void)out_size; (void)ws_size;
  const float* psi   = (const float*)d_in[0];
  const float* field = (const float*)d_in[1];
  const float* w     = (const float*)d_in[2];
  float* out = (float*)d_out;
  const int nrows = in_sizes[0] / kD;

  float* disp    = (float*)d_ws;        // [nrows]
  float* tot     = disp + nrows;        // [nrows]
  float* fmn     = tot + nrows;         // [nrows]
  float* fmx     = fmn + nrows;         // [nrows]
  float* rowc    = fmx + nrows;         // [nrows]
  float* scalars = rowc + nrows;        // [4]: gmaxf, cmin, cmax, gmaxt

  rowstats_kernel<<<nrows, kNT, 0, stream>>>(psi, field, disp, tot, fmn, fmx);
  globalred_kernel<<<1, 1024, 0, stream>>>(w, disp, tot, fmn, fmx, rowc, scalars, nrows);
  finalize_kernel<<<nrows, kNT, 0, stream>>>(field, w, rowc, scalars, out);
}